// GVPGraphEmbedding_77438260347483
// MI455X (gfx1250) — compile-verified
//
#include <hip/hip_runtime.h>
#include <math.h>

// Problem constants (match reference)
#define BB    4
#define LL    2048
#define KNN   30
#define NNODE (BB*LL)            // 8192
#define NEDGE (NNODE*KNN)        // 245760
#define NSF   1024
#define NVF   256
#define KPAD  264                // 263 padded to multiple of 4 for WMMA K=4
#define ESF   32

typedef float v2f __attribute__((ext_vector_type(2)));
typedef float v8f __attribute__((ext_vector_type(8)));

// d_out float offsets (outputs concatenated in return order: ns, nv, es, ev, ei)
#define OFF_NS 0
#define OFF_NV (OFF_NS + NNODE*NSF)
#define OFF_ES (OFF_NV + NNODE*NVF*3)
#define OFF_EV (OFF_ES + NEDGE*ESF)
#define OFF_EI (OFF_EV + NEDGE*3)

__device__ __forceinline__ float dot3(const float* a, const float* b) {
  return a[0]*b[0] + a[1]*b[1] + a[2]*b[2];
}
__device__ __forceinline__ void cross3(const float* a, const float* b, float* o) {
  o[0] = a[1]*b[2] - a[2]*b[1];
  o[1] = a[2]*b[0] - a[0]*b[2];
  o[2] = a[0]*b[1] - a[1]*b[0];
}
// t / sqrt(|t|^2 + 1e-8)  (matches _normalize; finite so nan_to_num is a no-op)
__device__ __forceinline__ void normz(float* v) {
  float inv = rsqrtf(dot3(v, v) + 1e-8f);
  v[0] *= inv; v[1] *= inv; v[2] *= inv;
}

// ---------------------------------------------------------------------------
// init: zero per-batch E_mean accumulators
// ---------------------------------------------------------------------------
__global__ void init_kernel(float* __restrict__ bsum) {
  if (threadIdx.x < 16) bsum[threadIdx.x] = 0.f;
}

// ---------------------------------------------------------------------------
// pad node_ws_w [263,1024] -> [264,1024] with a zero row (WMMA K multiple of 4)
// ---------------------------------------------------------------------------
__global__ void padB_kernel(const float* __restrict__ W, float* __restrict__ Bp) {
  int idx = blockIdx.x * blockDim.x + threadIdx.x;
  if (idx >= KPAD * NSF) return;
  int r = idx / NSF;
  Bp[idx] = (r < NVF + 7) ? W[idx] : 0.f;
}

// ---------------------------------------------------------------------------
// Per-node geometric features: dihedrals(6) + mask(1) -> s_in[:,0..6],
// orientations f/bk + sidechain -> nodev[n][3][3]
// ---------------------------------------------------------------------------
__global__ void node_feat_kernel(const float* __restrict__ coords,
                                 const unsigned char* __restrict__ coord_mask,
                                 float* __restrict__ s_in,
                                 float* __restrict__ nodev) {
  int n = blockIdx.x * blockDim.x + threadIdx.x;
  if (n >= NNODE) return;
  int b = n / LL, i = n % LL;
  const float* cb = coords + (size_t)b * LL * 9;   // flat atoms: cb[j*3+d], j=res*3+atom

  float cosd[3], sind[3];
  for (int s3 = 0; s3 < 3; ++s3) {
    int t = 3 * i + s3;
    float co = 1.f, si = 0.f;                      // padded entries: D=0
    if (t >= 1 && t <= 3 * LL - 3) {
      int j = t - 1;
      float u2[3], u1[3], u0[3];
      for (int d = 0; d < 3; ++d) {
        float p0 = cb[(j + 0) * 3 + d], p1 = cb[(j + 1) * 3 + d];
        float p2 = cb[(j + 2) * 3 + d], p3 = cb[(j + 3) * 3 + d];
        u2[d] = p1 - p0; u1[d] = p2 - p1; u0[d] = p3 - p2;
      }
      normz(u2); normz(u1); normz(u0);
      float n2[3], n1v[3];
      cross3(u2, u1, n2);  normz(n2);
      cross3(u1, u0, n1v); normz(n1v);
      float cD = dot3(n2, n1v);
      cD = fminf(fmaxf(cD, -1.f + 1e-7f), 1.f - 1e-7f);
      float sg = dot3(u2, n1v);
      float sgn = (sg > 0.f) ? 1.f : ((sg < 0.f) ? -1.f : 0.f);
      if (sgn == 0.f) { co = 1.f; si = 0.f; }      // D = 0*acos(..) = 0
      else { co = cD; si = sgn * sqrtf(fmaxf(1.f - cD * cD, 0.f)); }
    }
    cosd[s3] = co; sind[s3] = si;
  }
  float* sp = s_in + (size_t)n * KPAD;
  sp[0] = cosd[0]; sp[1] = cosd[1]; sp[2] = cosd[2];
  sp[3] = sind[0]; sp[4] = sind[1]; sp[5] = sind[2];
  sp[6] = coord_mask[n] ? 1.f : 0.f;
  sp[KPAD - 1] = 0.f;                              // K padding column

  // orientations + sidechain (vectors: 0=f, 1=bk, 2=sc)
  float xc[3], f[3] = {0, 0, 0}, bk[3] = {0, 0, 0};
  for (int d = 0; d < 3; ++d) xc[d] = cb[(i * 3 + 1) * 3 + d];
  if (i < LL - 1) { for (int d = 0; d < 3; ++d) f[d]  = cb[((i + 1) * 3 + 1) * 3 + d] - xc[d]; normz(f); }
  if (i > 0)      { for (int d = 0; d < 3; ++d) bk[d] = cb[((i - 1) * 3 + 1) * 3 + d] - xc[d]; normz(bk); }
  float nn[3], cc[3];
  for (int d = 0; d < 3; ++d) {
    nn[d] = cb[(i * 3 + 0) * 3 + d] - xc[d];
    cc[d] = cb[(i * 3 + 2) * 3 + d] - xc[d];
  }
  normz(cc); normz(nn);
  float bis[3], perp[3];
  for (int d = 0; d < 3; ++d) bis[d] = cc[d] + nn[d];
  normz(bis);
  cross3(cc, nn, perp); normz(perp);
  const float c13 = 0.57735026918962576f;          // sqrt(1/3)
  const float c23 = 0.81649658092772603f;          // sqrt(2/3)
  float* vp = nodev + (size_t)n * 9;
  for (int d = 0; d < 3; ++d) {
    vp[d]     = f[d];
    vp[3 + d] = bk[d];
    vp[6 + d] = -bis[d] * c13 - perp[d] * c23;
  }
}

// ---------------------------------------------------------------------------
// vh = einsum('id,ih->hd', v, wh); vn -> s_in[:,7+h]; vh stored as 3 d-planes
// ---------------------------------------------------------------------------
__global__ void vh_vn_kernel(const float* __restrict__ nodev,
                             const float* __restrict__ wh,
                             float* __restrict__ vh,
                             float* __restrict__ s_in) {
  int idx = blockIdx.x * blockDim.x + threadIdx.x;
  if (idx >= NNODE * NVF) return;
  int n = idx / NVF, h = idx % NVF;
  const float* vp = nodev + (size_t)n * 9;
  float w0 = wh[0 * NVF + h], w1 = wh[1 * NVF + h], w2 = wh[2 * NVF + h];
  float s2 = 0.f;
  for (int d = 0; d < 3; ++d) {
    float val = vp[d] * w0 + vp[3 + d] * w1 + vp[6 + d] * w2;
    vh[(size_t)d * NNODE * NVF + (size_t)n * NVF + h] = val;
    s2 += val * val;
  }
  s_in[(size_t)n * KPAD + 7 + h] = sqrtf(fmaxf(s2, 1e-8f));
}

// ---------------------------------------------------------------------------
// kNN: one block per node row; LDS D_adjust; 30 argmin rounds with JAX top_k
// lowest-index tie-break; accumulates per-batch E_mean numerator/denominator.
// ---------------------------------------------------------------------------
__global__ void knn_kernel(const float* __restrict__ coords,
                           const unsigned char* __restrict__ coord_mask,
                           const int* __restrict__ res_idx,
                           const unsigned char* __restrict__ padding_mask,
                           int* __restrict__ E_idx, float* __restrict__ E_dist,
                           float* __restrict__ bsum) {
  __shared__ float Dadj[LL];
  __shared__ float Dtru[LL];
  __shared__ float redv[256];
  __shared__ int   redi[256];
  __shared__ int   seli[KNN];
  __shared__ float seld[KNN];
  int n = blockIdx.x;
  int b = n / LL, i = n % LL;
  int t = threadIdx.x;
  const float* cb = coords + (size_t)b * LL * 9;
  float xi[3];
  for (int d = 0; d < 3; ++d) xi[d] = cb[(i * 3 + 1) * 3 + d];
  float cmi = coord_mask[n] ? 1.f : 0.f;
  float rmi = padding_mask[n] ? 0.f : 1.f;
  int   ri  = res_idx[n];
  for (int j = t; j < LL; j += 256) {
    float dx = xi[0] - cb[(j * 3 + 1) * 3 + 0];
    float dy = xi[1] - cb[(j * 3 + 1) * 3 + 1];
    float dz = xi[2] - cb[(j * 3 + 1) * 3 + 2];
    float dist = sqrtf(dx * dx + dy * dy + dz * dz + 1e-8f);
    float cmj = coord_mask[b * LL + j] ? 1.f : 0.f;
    float rmj = padding_mask[b * LL + j] ? 0.f : 1.f;
    float cm2 = cmi * cmj, rm2 = rmi * rmj;
    float D = cm2 * dist;
    int ad = abs(ri - res_idx[b * LL + j]);
    float cov = (ad <= 3) ? 1.f : 0.f;
    float Dseq = fabsf((float)(i - j));
    Dadj[j] = D * (1.f - cov) + (1.f - cm2) * (1e8f + Dseq * 1e6f) + (1.f - rm2) * 1e10f;
    Dtru[j] = D;
  }
  __syncthreads();
  for (int r = 0; r < KNN; ++r) {
    float best = 3.0e38f; int bidx = LL;
    for (int j = t; j < LL; j += 256) {            // strict < keeps lowest j per thread
      float v = Dadj[j];
      if (v < best) { best = v; bidx = j; }
    }
    redv[t] = best; redi[t] = bidx;
    __syncthreads();
    for (int s = 128; s > 0; s >>= 1) {
      if (t < s) {
        float v2 = redv[t + s]; int i2 = redi[t + s];
        if (v2 < redv[t] || (v2 == redv[t] && i2 < redi[t])) { redv[t] = v2; redi[t] = i2; }
      }
      __syncthreads();
    }
    if (t == 0) {
      int sel = redi[0];
      seli[r] = sel; seld[r] = Dtru[sel];
      Dadj[sel] = 3.4e38f;
      E_idx[(size_t)n * KNN + r]  = sel;
      E_dist[(size_t)n * KNN + r] = Dtru[sel];
    }
    __syncthreads();
  }
  if (t == 0) {                                    // E_mean accumulation (pre-infill E_vec * E_cm)
    float sx = 0, sy = 0, sz = 0, sc = 0;
    for (int r = 0; r < KNN; ++r) {
      int j = seli[r];
      float cm = (seld[r] < 5e7f) ? 1.f : 0.f;
      sx += cm * (xi[0] - cb[(j * 3 + 1) * 3 + 0]);
      sy += cm * (xi[1] - cb[(j * 3 + 1) * 3 + 1]);
      sz += cm * (xi[2] - cb[(j * 3 + 1) * 3 + 2]);
      sc += cm;
    }
    atomicAdd(&bsum[b * 4 + 0], sx);
    atomicAdd(&bsum[b * 4 + 1], sy);
    atomicAdd(&bsum[b * 4 + 2], sz);
    atomicAdd(&bsum[b * 4 + 3], sc);
  }
}

// ---------------------------------------------------------------------------
// f32 WMMA GEMM, C = A[M,K] * B[K,N] (+bias). One wave per 64x16 strip
// (4 stacked 16x16 tiles sharing each B fragment). V_WMMA_F32_16X16X4_F32.
// ISA layouts: A lane m=lane%16, K-halves {0,1}/{2,3}; C/D M = r + 8*(lane>=16).
// ---------------------------------------------------------------------------
__global__ void wmma_gemm_f32_k4(const float* __restrict__ A,
                                 const float* __restrict__ Bm,
                                 const float* __restrict__ bias,
                                 float* __restrict__ C,
                                 int M, int N, int K, int lda, int ldb, int ldc,
                                 long long sA, long long sB, long long sC) {
  const int lane = threadIdx.x & 31;
  const int wave = threadIdx.x >> 5;
  const int wavesPerBlk = blockDim.x >> 5;
  const int tilesN = N >> 4;
  const int supM   = M >> 6;                       // 64-row supertiles
  int wid = blockIdx.x * wavesPerBlk + wave;
  if (wid >= supM * tilesN) return;
  int sm = wid / tilesN;
  int tn = wid % tilesN;
  const float* Ap = A  + (long long)blockIdx.z * sA;
  const float* Bp = Bm + (long long)blockIdx.z * sB;
  float*       Cp = C  + (long long)blockIdx.z * sC;

  const int l15 = lane & 15;
  const int kh  = (lane < 16) ? 0 : 2;             // K half offset
  const int nn  = (tn << 4) + l15;                 // output column
  const int m0  = (sm << 6) + l15;                 // A row for subtile 0

  v8f acc0 = {0.f,0.f,0.f,0.f,0.f,0.f,0.f,0.f};
  v8f acc1 = acc0, acc2 = acc0, acc3 = acc0;

  for (int k0 = 0; k0 < K; k0 += 4) {
    v2f bf;
    bf.x = Bp[(long long)(k0 + kh    ) * ldb + nn];
    bf.y = Bp[(long long)(k0 + kh + 1) * ldb + nn];
    const float* ab = Ap + (long long)m0 * lda + k0 + kh;
    v2f a0, a1, a2, a3;
    a0.x = ab[0];                 a0.y = ab[1];
    a1.x = ab[(long long)16*lda]; a1.y = ab[(long long)16*lda + 1];
    a2.x = ab[(long long)32*lda]; a2.y = ab[(long long)32*lda + 1];
    a3.x = ab[(long long)48*lda]; a3.y = ab[(long long)48*lda + 1];
    acc0 = __builtin_amdgcn_wmma_f32_16x16x4_f32(false, a0, false, bf, (short)0, acc0, false, false);
    acc1 = __builtin_amdgcn_wmma_f32_16x16x4_f32(false, a1, false, bf, (short)0, acc1, false, false);
    acc2 = __builtin_amdgcn_wmma_f32_16x16x4_f32(false, a2, false, bf, (short)0, acc2, false, false);
    acc3 = __builtin_amdgcn_wmma_f32_16x16x4_f32(false, a3, false, bf, (short)0, acc3, false, false);
  }
  const int mb = (sm << 6) + ((lane < 16) ? 0 : 8);
  const float bv = bias ? bias[nn] : 0.f;
  for (int r = 0; r < 8; ++r) {
    Cp[(long long)(mb + r     ) * ldc + nn] = acc0[r] + bv;
    Cp[(long long)(mb + r + 16) * ldc + nn] = acc1[r] + bv;
    Cp[(long long)(mb + r + 32) * ldc + nn] = acc2[r] + bv;
    Cp[(long long)(mb + r + 48) * ldc + nn] = acc3[r] + bv;
  }
}

// ---------------------------------------------------------------------------
// ns LayerNorm + confidence-RBF projection, in place on d_out ns region
// ---------------------------------------------------------------------------
__global__ void ln_ns_kernel(float* __restrict__ ns,
                             const float* __restrict__ g, const float* __restrict__ bt,
                             const float* __restrict__ confidence,
                             const float* __restrict__ conf_w,
                             const float* __restrict__ conf_b) {
  __shared__ float s1[256], s2[256];
  int n = blockIdx.x, t = threadIdx.x;
  float* row = ns + (size_t)n * NSF;
  float x[4], a = 0.f, q = 0.f;
  for (int k = 0; k < 4; ++k) { x[k] = row[t + k * 256]; a += x[k]; q += x[k] * x[k]; }
  s1[t] = a; s2[t] = q;
  __syncthreads();
  for (int s = 128; s > 0; s >>= 1) {
    if (t < s) { s1[t] += s1[t + s]; s2[t] += s2[t + s]; }
    __syncthreads();
  }
  float mu   = s1[0] * (1.f / NSF);
  float var  = s2[0] * (1.f / NSF) - mu * mu;
  float rinv = rsqrtf(var + 1e-4f);
  float cv = confidence[n];
  float w16[16];
  for (int r = 0; r < 16; ++r) {
    float z = (cv - (float)r * (1.f / 15.f)) * 16.f;   // centers linspace(0,1,16), std=1/16
    w16[r] = expf(-z * z);
  }
  for (int k = 0; k < 4; ++k) {
    int c = t + k * 256;
    float dv = 0.f;
    for (int r = 0; r < 16; ++r) dv += w16[r] * conf_w[r * NSF + c];
    row[c] = (x[k] - mu) * rinv * g[c] + bt[c] + dv + conf_b[c];
  }
}

// ---------------------------------------------------------------------------
// nv vector LayerNorm: v / sqrt(mean_h max(|v_h|^2,1e-8)); interleave to [n,h,3]
// ---------------------------------------------------------------------------
__global__ void ln_nv_kernel(const float* __restrict__ nvpre, float* __restrict__ outv) {
  __shared__ float s1[256];
  int n = blockIdx.x, h = threadIdx.x;
  float x = nvpre[0 * (size_t)NNODE * NVF + (size_t)n * NVF + h];
  float y = nvpre[1 * (size_t)NNODE * NVF + (size_t)n * NVF + h];
  float z = nvpre[2 * (size_t)NNODE * NVF + (size_t)n * NVF + h];
  s1[h] = fmaxf(x * x + y * y + z * z, 1e-8f);
  __syncthreads();
  for (int s = 128; s > 0; s >>= 1) {
    if (h < s) s1[h] += s1[h + s];
    __syncthreads();
  }
  float sc = rsqrtf(s1[0] * (1.f / NVF));
  float* o = outv + ((size_t)n * NVF + h) * 3;
  o[0] = x * sc; o[1] = y * sc; o[2] = z * sc;
}

// ---------------------------------------------------------------------------
// Edge features + edge GVP + edge LN + edge_index; one thread per edge
// ---------------------------------------------------------------------------
__global__ void edge_kernel(const float* __restrict__ coords,
                            const unsigned char* __restrict__ coord_mask,
                            const int* __restrict__ res_idx,
                            const int* __restrict__ E_idx,
                            const float* __restrict__ E_dist,
                            const float* __restrict__ bsum,
                            const float* __restrict__ ewh, const float* __restrict__ ews_w,
                            const float* __restrict__ ews_b, const float* __restrict__ ewv,
                            const float* __restrict__ elg, const float* __restrict__ elb,
                            float* __restrict__ out) {
  __shared__ float sW[35 * ESF];
  __shared__ float sB[ESF], sG[ESF], sBt[ESF];
  for (int q = threadIdx.x; q < 35 * ESF; q += blockDim.x) sW[q] = ews_w[q];
  if (threadIdx.x < ESF) {
    sB[threadIdx.x] = ews_b[threadIdx.x];
    sG[threadIdx.x] = elg[threadIdx.x];
    sBt[threadIdx.x] = elb[threadIdx.x];
  }
  __syncthreads();
  int e = blockIdx.x * blockDim.x + threadIdx.x;
  if (e >= NEDGE) return;
  int n = e / KNN;
  int b = n / LL, i = n % LL;
  int j = E_idx[e];
  float dist = E_dist[e];
  bool Ecm = dist < 5e7f;
  bool Erm = dist < 5e9f;
  bool invalid = !(Ecm && Erm);
  const float* cb = coords + (size_t)b * LL * 9;
  float ev[3];
  float cnt = bsum[b * 4 + 3];
  for (int d = 0; d < 3; ++d) {
    float diff = cb[(i * 3 + 1) * 3 + d] - cb[(j * 3 + 1) * 3 + d];
    float mean = bsum[b * 4 + d] / cnt;
    ev[d] = Ecm ? diff : mean;                    // E_vec*cm + E_mean*(1-cm)
  }
  normz(ev);
  float sfeat[35];
  for (int r = 0; r < 16; ++r) {                  // RBF over [0,20], 16 bins, std=1.25
    float zz = (dist - 20.f * (float)r * (1.f / 15.f)) * (1.f / 1.25f);
    sfeat[r] = expf(-zz * zz);
  }
  float dd = fminf(fmaxf((float)(res_idx[n] - res_idx[b * LL + j]), -32.f), 32.f);
  for (int m = 0; m < 8; ++m) {
    float fr = expf((float)(2 * m) * (-9.210340371976184f / 16.f)); // 10000^(-m/8)
    float ang = dd * fr;
    sfeat[16 + m] = cosf(ang);
    sfeat[24 + m] = sinf(ang);
  }
  sfeat[32] = coord_mask[n] ? 0.f : 1.f;          // ~cm_src
  sfeat[33] = coord_mask[b * LL + j] ? 0.f : 1.f; // ~cm_dest
  float whs = ewh[0], wvs = ewv[0];
  float vh[3] = {ev[0] * whs, ev[1] * whs, ev[2] * whs};
  sfeat[34] = sqrtf(fmaxf(vh[0] * vh[0] + vh[1] * vh[1] + vh[2] * vh[2], 1e-8f));
  float so[ESF], mu = 0.f;
  for (int c = 0; c < ESF; ++c) {
    float acc = sB[c];
    for (int r = 0; r < 35; ++r) acc += sfeat[r] * sW[r * ESF + c];
    so[c] = acc; mu += acc;
  }
  mu *= (1.f / ESF);
  float var = 0.f;
  for (int c = 0; c < ESF; ++c) { float dv = so[c] - mu; var += dv * dv; }
  var *= (1.f / ESF);
  float rinv = rsqrtf(var + 1e-4f);
  float vo[3] = {vh[0] * wvs, vh[1] * wvs, vh[2] * wvs};
  float vnorm = rsqrtf(fmaxf(vo[0] * vo[0] + vo[1] * vo[1] + vo[2] * vo[2], 1e-8f));
  float* oes = out + OFF_ES + (size_t)e * ESF;
  for (int c = 0; c < ESF; ++c)
    oes[c] = invalid ? 0.f : ((so[c] - mu) * rinv * sG[c] + sBt[c]);
  float* oev = out + OFF_EV + (size_t)e * 3;
  for (int d = 0; d < 3; ++d) oev[d] = invalid ? 0.f : vo[d] * vnorm;
  int* oei = (int*)(out + OFF_EI);
  oei[e]         = invalid ? -1 : n;              // global src
  oei[NEDGE + e] = invalid ? -1 : (b * LL + j);   // global dest
}

// ---------------------------------------------------------------------------
extern "C" void kernel_launch(void* const* d_in, const int* in_sizes, int n_in,
                              void* d_out, int out_size, void* d_ws, size_t ws_size,
                              hipStream_t stream) {
  (void)in_sizes; (void)n_in; (void)out_size; (void)ws_size;
  const float*         coords       = (const float*)d_in[0];
  const unsigned char* coord_mask   = (const unsigned char*)d_in[1];   // numpy bool = 1 byte
  const int*           res_idx      = (const int*)d_in[2];
  const unsigned char* padding_mask = (const unsigned char*)d_in[3];
  const float*         confidence   = (const float*)d_in[4];
  const float*         node_wh      = (const float*)d_in[5];
  const float*         node_ws_w    = (const float*)d_in[6];
  const float*         node_ws_b    = (const float*)d_in[7];
  const float*         node_wv      = (const float*)d_in[8];
  const float*         node_ln_g    = (const float*)d_in[9];
  const float*         node_ln_b    = (const float*)d_in[10];
  const float*         edge_wh      = (const float*)d_in[11];
  const float*         edge_ws_w    = (const float*)d_in[12];
  const float*         edge_ws_b    = (const float*)d_in[13];
  const float*         edge_wv      = (const float*)d_in[14];
  const float*         edge_ln_g    = (const float*)d_in[15];
  const float*         edge_ln_b    = (const float*)d_in[16];
  const float*         conf_w       = (const float*)d_in[17];
  const float*         conf_b       = (const float*)d_in[18];
  float* out = (float*)d_out;
  char*  ws  = (char*)d_ws;

  size_t o = 0;
  float* ws_sin   = (float*)(ws + o); o += (size_t)NNODE * KPAD * 4;
  float* ws_vh    = (float*)(ws + o); o += (size_t)3 * NNODE * NVF * 4;
  float* ws_nodev = (float*)(ws + o); o += (size_t)NNODE * 9 * 4;
  float* ws_nvpre = (float*)(ws + o); o += (size_t)3 * NNODE * NVF * 4;
  int*   ws_eidx  = (int*)  (ws + o); o += (size_t)NEDGE * 4;
  float* ws_edist = (float*)(ws + o); o += (size_t)NEDGE * 4;
  float* ws_bsum  = (float*)(ws + o); o += 256;
  float* ws_Bpad  = (float*)(ws + o); o += (size_t)KPAD * NSF * 4;

  init_kernel<<<1, 32, 0, stream>>>(ws_bsum);
  padB_kernel<<<(KPAD * NSF + 255) / 256, 256, 0, stream>>>(node_ws_w, ws_Bpad);
  node_feat_kernel<<<NNODE / 256, 256, 0, stream>>>(coords, coord_mask, ws_sin, ws_nodev);
  vh_vn_kernel<<<(NNODE * NVF) / 256, 256, 0, stream>>>(ws_nodev, node_wh, ws_vh, ws_sin);
  knn_kernel<<<NNODE, 256, 0, stream>>>(coords, coord_mask, res_idx, padding_mask,
                                        ws_eidx, ws_edist, ws_bsum);
  // ns = s_in[8192,264] @ Bpad[264,1024] + bias  (written straight into d_out)
  {
    int waves = (NNODE / 64) * (NSF / 16);               // 8192 waves
    wmma_gemm_f32_k4<<<dim3(waves / 8, 1, 1), 256, 0, stream>>>(
        ws_sin, ws_Bpad, node_ws_b, out + OFF_NS,
        NNODE, NSF, KPAD, KPAD, NSF, NSF, 0, 0, 0);
  }
  // nv planes: vh_d[8192,256] @ wv[256,256], d=0..2 via grid.z
  {
    int waves = (NNODE / 64) * (NVF / 16);               // 2048 waves per plane
    wmma_gemm_f32_k4<<<dim3(waves / 8, 1, 3), 256, 0, stream>>>(
        ws_vh, node_wv, nullptr, ws_nvpre,
        NNODE, NVF, NVF, NVF, NVF, NVF,
        (long long)NNODE * NVF, 0, (long long)NNODE * NVF);
  }
  ln_ns_kernel<<<NNODE, 256, 0, stream>>>(out + OFF_NS, node_ln_g, node_ln_b,
                                          confidence, conf_w, conf_b);
  ln_nv_kernel<<<NNODE, 256, 0, stream>>>(ws_nvpre, out + OFF_NV);
  edge_kernel<<<NEDGE / 256, 256, 0, stream>>>(coords, coord_mask, res_idx,
                                               ws_eidx, ws_edist, ws_bsum,
                                               edge_wh, edge_ws_w, edge_ws_b, edge_wv,
                                               edge_ln_g, edge_ln_b, out);
}